// ExampleTiedDropout_48473000903475
// MI455X (gfx1250) — compile-verified
//
#include <hip/hip_runtime.h>
#include <hip/hip_bf16.h>
#include <stdint.h>

// ---------------------------------------------------------------------------
// Threefry-2x32 (exact JAX semantics: 20 rounds, 5 key injections)
// ---------------------------------------------------------------------------
__device__ __forceinline__ uint32_t rotl32(uint32_t v, int n) {
  return (v << n) | (v >> (32 - n));
}

__device__ __forceinline__ void threefry2x32(uint32_t k0, uint32_t k1,
                                             uint32_t x0, uint32_t x1,
                                             uint32_t& o0, uint32_t& o1) {
  const uint32_t kx = k0 ^ k1 ^ 0x1BD11BDAu;
  uint32_t y0 = x0 + k0;
  uint32_t y1 = x1 + k1;
  const int ra0 = 13, ra1 = 15, ra2 = 26, ra3 = 6;
  const int rb0 = 17, rb1 = 29, rb2 = 16, rb3 = 24;

#define TF_ROUND(r) do { y0 += y1; y1 = rotl32(y1, (r)); y1 ^= y0; } while (0)
  TF_ROUND(ra0); TF_ROUND(ra1); TF_ROUND(ra2); TF_ROUND(ra3);
  y0 += k1;  y1 += kx + 1u;
  TF_ROUND(rb0); TF_ROUND(rb1); TF_ROUND(rb2); TF_ROUND(rb3);
  y0 += kx;  y1 += k0 + 2u;
  TF_ROUND(ra0); TF_ROUND(ra1); TF_ROUND(ra2); TF_ROUND(ra3);
  y0 += k0;  y1 += k1 + 3u;
  TF_ROUND(rb0); TF_ROUND(rb1); TF_ROUND(rb2); TF_ROUND(rb3);
  y0 += k1;  y1 += kx + 4u;
  TF_ROUND(ra0); TF_ROUND(ra1); TF_ROUND(ra2); TF_ROUND(ra3);
  y0 += kx;  y1 += k0 + 5u;
#undef TF_ROUND
  o0 = y0;
  o1 = y1;
}

// JAX uniform-from-bits: bitcast((bits >> 9) | 0x3f800000) - 1.0f  in [0,1)
__device__ __forceinline__ float unit_from_bits(uint32_t bits) {
  return __uint_as_float((bits >> 9) | 0x3f800000u) - 1.0f;
}

// ---------------------------------------------------------------------------
// Problem geometry (from the reference)
// ---------------------------------------------------------------------------
#define B_EX     64
#define C_CH     512
#define HW       1024          // 32*32 floats per (b,c) slab
#define K_FIXED  102           // int(0.2 * 512)
#define N_DROP   410           // 512 - 102
#define N_PAIR   205           // N_DROP / 2 (JAX even split)
#define BASE_K0  0u
#define BASE_K1  1234u
#define P_MEM    0.1f

// ---------------------------------------------------------------------------
// Kernel 1: build mask[B_EX][C_CH] in workspace (one block per example)
// ---------------------------------------------------------------------------
__global__ __launch_bounds__(256)
void etd_mask_kernel(const int* __restrict__ idx, float* __restrict__ mask) {
  const int b = blockIdx.x;
  const uint32_t e = (uint32_t)idx[b];

  // fold_in(key(1234), e) == threefry2x32(key=(0,1234), msg=(hi(e)=0, e))
  uint32_t fk0, fk1;
  threefry2x32(BASE_K0, BASE_K1, 0u, e, fk0, fk1);

  float* mrow = mask + b * C_CH;
  const int t = threadIdx.x;

  if (t < N_PAIR) {
    // JAX even-length split: pair t -> bits for positions t and t+205
    uint32_t b0, b1;
    threefry2x32(fk0, fk1, (uint32_t)t, (uint32_t)(t + N_PAIR), b0, b1);
    mrow[K_FIXED + t]          = (unit_from_bits(b0) < P_MEM) ? 1.0f : 0.0f;
    mrow[K_FIXED + N_PAIR + t] = (unit_from_bits(b1) < P_MEM) ? 1.0f : 0.0f;
  } else if (t < N_PAIR + K_FIXED) {
    mrow[t - N_PAIR] = 1.0f;   // always-kept channels [0, 102)
  }
}

// ---------------------------------------------------------------------------
// Kernel 2: streaming apply via CDNA5 async global->LDS pipeline
//   one wave32 per (b,c) slab of 1024 floats = 8 tiles of (32 lanes x float4)
// ---------------------------------------------------------------------------
#define WAVES_PER_BLOCK 8
#define TILES           8      // tiles per slab
#define DEPTH           4      // async pipeline depth (512B tiles in flight)

__device__ __forceinline__ void async_load_b128(uint32_t lds_byte_addr,
                                                const void* gaddr) {
  // gfx1250 VGLOBAL async op: copies 16 B per lane from global memory into LDS and counts on ASYNCcnt
  asm volatile("global_load_async_to_lds_b128 %0, %1, off"
               :: "v"(lds_byte_addr), "v"(gaddr)
               : "memory");
}

__device__ __forceinline__ void wait_async_le(int n) {
  // n is compile-time constant after unrolling; dead branches fold away.
  switch (n) {
    case 0:  asm volatile("s_wait_asynccnt 0" ::: "memory"); break;
    case 1:  asm volatile("s_wait_asynccnt 1" ::: "memory"); break;
    case 2:  asm volatile("s_wait_asynccnt 2" ::: "memory"); break;
    default: asm volatile("s_wait_asynccnt 3" ::: "memory"); break;
  }
}

__global__ __launch_bounds__(WAVES_PER_BLOCK * 32)
void etd_apply_kernel(const float* __restrict__ X,
                      const float* __restrict__ mask,
                      float* __restrict__ out) {
  __shared__ float4 lbuf[WAVES_PER_BLOCK * DEPTH * 32];   // 16 KB / WG

  const int lane = threadIdx.x & 31;
  const int w    = threadIdx.x >> 5;
  const int slab = blockIdx.x * WAVES_PER_BLOCK + w;      // b*512 + c

  // wave-uniform mask value -> force into SGPR
  float m = mask[slab];
  m = __int_as_float(__builtin_amdgcn_readfirstlane(__float_as_int(m)));

  const float4* __restrict__ src = (const float4*)(X   + (size_t)slab * HW);
  float4* __restrict__       dst = (float4*)      (out + (size_t)slab * HW);

  float4* buf = &lbuf[w * (DEPTH * 32)];
  const uint32_t lds_base =
      (uint32_t)(uintptr_t)(buf + lane);                  // lane slot, stage 0
  const uint32_t stage_stride = 32u * sizeof(float4);     // 512 B per stage

  // prologue: fill the async pipeline
#pragma unroll
  for (int s = 0; s < DEPTH; ++s)
    async_load_b128(lds_base + (uint32_t)s * stage_stride, src + s * 32 + lane);

#pragma unroll
  for (int t = 0; t < TILES; ++t) {
    // wait until tile t has landed in LDS (in-order async completion)
    const int rem = TILES - 1 - t;
    wait_async_le(rem < (DEPTH - 1) ? rem : (DEPTH - 1));

    float4 v = buf[(t % DEPTH) * 32 + lane];              // ds_load_b128
    float4 y = make_float4(v.x * m, v.y * m, v.z * m, v.w * m);

    // ensure the LDS read retired before the async engine overwrites the stage
    asm volatile("s_wait_dscnt 0" ::: "memory");
    if (t + DEPTH < TILES)
      async_load_b128(lds_base + (uint32_t)(t % DEPTH) * stage_stride,
                      src + (t + DEPTH) * 32 + lane);

    dst[t * 32 + lane] = y;                               // global_store_b128
  }
}

// ---------------------------------------------------------------------------
// Launch wrapper
// ---------------------------------------------------------------------------
extern "C" void kernel_launch(void* const* d_in, const int* in_sizes, int n_in,
                              void* d_out, int out_size, void* d_ws, size_t ws_size,
                              hipStream_t stream) {
  const float* X   = (const float*)d_in[0];   // [64,512,32,32] f32
  const int*   idx = (const int*)  d_in[1];   // [64] i32
  float* out  = (float*)d_out;                // [64,512,32,32] f32
  float* mask = (float*)d_ws;                 // [64*512] f32 scratch

  // 1) per-example channel masks (exact JAX Threefry reproduction)
  etd_mask_kernel<<<B_EX, 256, 0, stream>>>(idx, mask);

  // 2) bandwidth-bound apply: 32768 slabs, one wave32 each,
  //    depth-4 async global->LDS pipeline per wave
  const int blocks = (B_EX * C_CH) / WAVES_PER_BLOCK;     // 4096
  etd_apply_kernel<<<blocks, WAVES_PER_BLOCK * 32, 0, stream>>>(X, mask, out);
}